// relativeloss_60541859005001
// MI455X (gfx1250) — compile-verified
//
#include <hip/hip_runtime.h>
#include <cstdint>

#define NRPTS 5000
#define IMG_W 640
#define NBLOCKS 1024
#define BLOCK 256

constexpr long long P_LL = (long long)NRPTS * (NRPTS - 1) / 2;

// -------- Kernel 1: gather z[i] = x[centers[i].row * W + centers[i].col] ----
__global__ __launch_bounds__(BLOCK) void gather_z_kernel(
    const float* __restrict__ x, const int* __restrict__ centers,
    float* __restrict__ z) {
  int i = blockIdx.x * BLOCK + threadIdx.x;
  if (i < NRPTS) {
    int r = centers[2 * i + 0];
    int c = centers[2 * i + 1];
    z[i] = x[r * IMG_W + c];
  }
}

// -------- Kernel 2: pairwise loss, z staged into LDS via async DMA ----------
__global__ __launch_bounds__(BLOCK) void pair_loss_kernel(
    const float* __restrict__ z, const float* __restrict__ gt,
    float* __restrict__ partials) {
  __shared__ __align__(16) float s_z[NRPTS];   // 20 000 B of the 320 KB WGP LDS
  __shared__ float s_red[BLOCK];
  const int tid = threadIdx.x;

  // Stage the whole z table into LDS with the CDNA5 async data mover.
  // 5000 floats = 1250 x b128 transfers, tracked by ASYNCcnt.
  {
    const uint32_t lds_base = (uint32_t)(uintptr_t)(&s_z[0]);
    for (int k = tid; k < NRPTS / 4; k += BLOCK) {
      uint64_t ga = (uint64_t)(uintptr_t)(z + 4 * k);
      uint32_t la = lds_base + 16u * (uint32_t)k;
      asm volatile("global_load_async_to_lds_b128 %0, %1, off"
                   :: "v"(la), "v"(ga) : "memory");
    }
    asm volatile("s_wait_asynccnt 0" ::: "memory");
  }
  __syncthreads();

  float acc = 0.0f;
  // Blocks stride over rows i; lanes stride over j -> gt reads are coalesced.
  for (int i = blockIdx.x; i < NRPTS - 1; i += NBLOCKS) {
    const float zi = s_z[i];
    // linear pair index of (i, j) is base + (j - i - 1)
    const int base = i * (NRPTS - 1) - (i * (i - 1)) / 2;
    const float* __restrict__ grow = gt + (base - (i + 1));
    for (int j = i + 1 + tid; j < NRPTS; j += BLOCK) {
      float g = grow[j];          // coalesced: address = base + lane stride
      float d = zi - s_z[j];      // LDS hit
      float t = -g * d;
      // gt in {-1,0,1}; |d| ~ N(0,2) so exp() cannot overflow here.
      float sp = __logf(1.0f + __expf(t));   // v_log_f32 / v_exp_f32
      acc += (g == 0.0f) ? (d * d) : sp;
    }
  }

  // Deterministic block reduction (fixed tree order), then one slot per block.
  s_red[tid] = acc;
  __syncthreads();
  for (int s = BLOCK / 2; s > 0; s >>= 1) {
    if (tid < s) s_red[tid] += s_red[tid + s];
    __syncthreads();
  }
  if (tid == 0) partials[blockIdx.x] = s_red[0];
}

// -------- Kernel 3: deterministic final reduction + scale by 1/P ------------
__global__ __launch_bounds__(BLOCK) void finalize_kernel(
    const float* __restrict__ partials, float* __restrict__ out) {
  __shared__ float s_red[BLOCK];
  const int tid = threadIdx.x;
  float a = 0.0f;
  for (int k = tid; k < NBLOCKS; k += BLOCK) a += partials[k];  // fixed order
  s_red[tid] = a;
  __syncthreads();
  for (int s = BLOCK / 2; s > 0; s >>= 1) {
    if (tid < s) s_red[tid] += s_red[tid + s];
    __syncthreads();
  }
  if (tid == 0) out[0] = s_red[0] * (float)(1.0 / (double)P_LL);
}

extern "C" void kernel_launch(void* const* d_in, const int* in_sizes, int n_in,
                              void* d_out, int out_size, void* d_ws, size_t ws_size,
                              hipStream_t stream) {
  const float* x       = (const float*)d_in[0];  // (H, W) float32
  const float* gt      = (const float*)d_in[1];  // (P,)   float32 in {-1,0,1}
  const int*   centers = (const int*)d_in[2];    // (NR,2) int32
  float* out = (float*)d_out;

  // Workspace layout: [0, 20000) z table, [20480, 20480+4*NBLOCKS) partials.
  float* z        = (float*)d_ws;
  float* partials = (float*)((char*)d_ws + 20480);

  gather_z_kernel<<<(NRPTS + BLOCK - 1) / BLOCK, BLOCK, 0, stream>>>(x, centers, z);
  pair_loss_kernel<<<NBLOCKS, BLOCK, 0, stream>>>(z, gt, partials);
  finalize_kernel<<<1, BLOCK, 0, stream>>>(partials, out);
}